// QModel_65481071397511
// MI455X (gfx1250) — compile-verified
//
#include <hip/hip_runtime.h>
#include <hip/hip_bf16.h>

typedef float v2f __attribute__((ext_vector_type(2)));
typedef float v8f __attribute__((ext_vector_type(8)));

#define B_DIM   32
#define S_LEN   512
#define N_SUB   4
#define T1      128   // threads per block in kernel 1 (S_LEN / N_SUB)

// ---------------------------------------------------------------------------
// Kernel 1: analytic quantum sim + per-batch partial pooling.
// e[n][0]   = prod_{j=1..7} cos(x[n,j]+p[h,j])
// e[n][w>0] = prod_{j=0..w} cos(x[n,j]+p[h,j])
// summed over 4 heads; partial-summed over 128 samples per block.
// ---------------------------------------------------------------------------
__global__ void __launch_bounds__(T1)
qsim_pool_kernel(const float* __restrict__ x, const float* __restrict__ rxp,
                 float* __restrict__ part)
{
    const int b   = blockIdx.x;       // batch
    const int sub = blockIdx.y;       // sub-chunk of S
    const int tid = threadIdx.x;
    const int s   = sub * T1 + tid;
    const int n   = b * S_LEN + s;    // sample index

    // trainable params (broadcast, L2/L0 cached)
    float P[4][8];
#pragma unroll
    for (int h = 0; h < 4; ++h)
#pragma unroll
        for (int j = 0; j < 8; ++j)
            P[h][j] = rxp[h * 8 + j];

    // 8 contiguous floats per sample, 32B aligned
    const float4* xv = reinterpret_cast<const float4*>(x + (size_t)n * 8);
    const float4 xa = xv[0], xb = xv[1];
    const float xs[8] = {xa.x, xa.y, xa.z, xa.w, xb.x, xb.y, xb.z, xb.w};

    float e[8];
#pragma unroll
    for (int w = 0; w < 8; ++w) e[w] = 0.0f;

#pragma unroll
    for (int h = 0; h < 4; ++h) {
        float c[8];
#pragma unroll
        for (int j = 0; j < 8; ++j)
            c[j] = __cosf(xs[j] + P[h][j]);
        float r = c[0];
#pragma unroll
        for (int w = 1; w < 8; ++w) { r *= c[w]; e[w] += r; }
        float q = c[1];
#pragma unroll
        for (int j = 2; j < 8; ++j) q *= c[j];
        e[0] += q;
    }

    // deterministic block tree reduction over 8 components
    __shared__ float red[T1][8];
#pragma unroll
    for (int w = 0; w < 8; ++w) red[tid][w] = e[w];
    __syncthreads();
#pragma unroll
    for (int st = T1 / 2; st >= 1; st >>= 1) {
        if (tid < st) {
#pragma unroll
            for (int w = 0; w < 8; ++w) red[tid][w] += red[tid + st][w];
        }
        __syncthreads();
    }
    if (tid < 8)
        part[(b * N_SUB + sub) * 8 + tid] = red[0][tid];
}

// ---------------------------------------------------------------------------
// Kernel 2: single wave32 block.
//   E[32][8]  = pooled mean of e (folding mean over heads and S)
//   A = W1 @ Wc (32x8), bb = W1 @ bc + b1
//   H = relu(E @ A^T + bb) via V_WMMA_F32_16X16X4_F32 (2x2 tiles, 2 K-steps)
//   y[b] = sum_i H[b][i] * W2[i] + b2
// ---------------------------------------------------------------------------
__global__ void __launch_bounds__(32)
epilogue_kernel(const float* __restrict__ part,
                const float* __restrict__ Wc, const float* __restrict__ bc,
                const float* __restrict__ W1, const float* __restrict__ b1,
                const float* __restrict__ W2, const float* __restrict__ b2,
                float* __restrict__ out)
{
    __shared__ float E[32][8];
    __shared__ float A[32][8];
    __shared__ float bb[32];
    __shared__ float HW[32][32];

    const int lane = threadIdx.x;   // 0..31, wave32, EXEC all ones

    // pooled mean e per batch row
#pragma unroll
    for (int k = 0; k < 8; ++k) {
        float s = 0.0f;
#pragma unroll
        for (int sub = 0; sub < N_SUB; ++sub)
            s += part[(lane * N_SUB + sub) * 8 + k];
        E[lane][k] = s * (1.0f / (4.0f * (float)S_LEN));   // /heads /S
    }

    // A = W1 @ Wc ; bb = W1 @ bc + b1   (lane = hidden row i)
#pragma unroll
    for (int k = 0; k < 8; ++k) {
        float a = 0.0f;
#pragma unroll
        for (int m = 0; m < 8; ++m)
            a += W1[lane * 8 + m] * Wc[m * 8 + k];
        A[lane][k] = a;
    }
    {
        float ab = b1[lane];
#pragma unroll
        for (int m = 0; m < 8; ++m) ab += W1[lane * 8 + m] * bc[m];
        bb[lane] = ab;
    }
    __syncthreads();

    const int lo = lane & 15;
    const int hi = lane >> 4;

    v8f acc[2][2] = {};
#pragma unroll
    for (int kt = 0; kt < 2; ++kt) {
        // A-frag 16x4 f32: lane half selects K pair {0,1} vs {2,3}
        const int kb = kt * 4 + 2 * hi;
        v2f af[2], bf[2];
#pragma unroll
        for (int t = 0; t < 2; ++t) {
            af[t].x = E[lo + 16 * t][kb];     // M rows = batch
            af[t].y = E[lo + 16 * t][kb + 1];
            bf[t].x = A[lo + 16 * t][kb];     // B[k][n] = A[n][k]
            bf[t].y = A[lo + 16 * t][kb + 1];
        }
#pragma unroll
        for (int mt = 0; mt < 2; ++mt)
#pragma unroll
            for (int nt = 0; nt < 2; ++nt)
                acc[mt][nt] = __builtin_amdgcn_wmma_f32_16x16x4_f32(
                    false, af[mt], false, bf[nt],
                    (short)0, acc[mt][nt], false, false);
    }

    // bias + relu + scale by W2, scatter to LDS (each (b,i) written once)
    float w2l[2];
    w2l[0] = W2[lo];
    w2l[1] = W2[lo + 16];
#pragma unroll
    for (int mt = 0; mt < 2; ++mt)
#pragma unroll
        for (int nt = 0; nt < 2; ++nt)
#pragma unroll
            for (int v = 0; v < 8; ++v) {
                const int bi = v + 8 * hi + 16 * mt;   // batch row (D layout)
                const int ii = lo + 16 * nt;           // hidden col
                float hv = acc[mt][nt][v] + bb[ii];
                hv = fmaxf(hv, 0.0f);
                HW[bi][ii] = hv * w2l[nt];
            }
    __syncthreads();

    float y = b2[0];
#pragma unroll
    for (int i = 0; i < 32; ++i) y += HW[lane][i];
    out[lane] = y;
}

extern "C" void kernel_launch(void* const* d_in, const int* in_sizes, int n_in,
                              void* d_out, int out_size, void* d_ws, size_t ws_size,
                              hipStream_t stream) {
    const float* x   = (const float*)d_in[0];
    const float* rxp = (const float*)d_in[1];
    const float* Wc  = (const float*)d_in[2];
    const float* bc  = (const float*)d_in[3];
    const float* W1  = (const float*)d_in[4];
    const float* b1  = (const float*)d_in[5];
    const float* W2  = (const float*)d_in[6];
    const float* b2  = (const float*)d_in[7];
    float* out  = (float*)d_out;
    float* part = (float*)d_ws;   // [32][N_SUB][8] floats = 4 KB

    dim3 grid(B_DIM, N_SUB);
    qsim_pool_kernel<<<grid, T1, 0, stream>>>(x, rxp, part);
    epilogue_kernel<<<1, 32, 0, stream>>>(part, Wc, bc, W1, b1, W2, b2, out);
}